// RGCN_65609920414359
// MI455X (gfx1250) — compile-verified
//
#include <hip/hip_runtime.h>
#include <hip/hip_bf16.h>

#define N_NODES 100000
#define N_EDGES 1000000
#define D 64
#define N_REL 100
#define N_TRIPLES (1024 * 33)

typedef float v2f __attribute__((ext_vector_type(2)));
typedef float v8f __attribute__((ext_vector_type(8)));

// ---------------------------------------------------------------------------
// Zero-fill (agg buffers must start at 0 every call; d_ws is poisoned 0xAA).
// ---------------------------------------------------------------------------
__global__ void rgcn_zero_kernel(float4* __restrict__ p, int n4) {
    int i = blockIdx.x * blockDim.x + threadIdx.x;
    if (i < n4) p[i] = make_float4(0.f, 0.f, 0.f, 0.f);
}

// ---------------------------------------------------------------------------
// H = (RELU_IN ? relu(X) : X) @ W + bias     [N x 64] @ [64 x 64]
// One wave per 16-row strip; 4 WMMA accumulators cover the 64 output cols.
// Bias is folded into the accumulator seed (D = A*B + C chain carries it).
// Uses V_WMMA_F32_16X16X4_F32 (fp32 end-to-end, matches reference).
// ---------------------------------------------------------------------------
template <bool RELU_IN>
__global__ void rgcn_gemm_wmma(const float* __restrict__ X,
                               const float* __restrict__ W,
                               const float* __restrict__ bias,
                               float* __restrict__ H,
                               int n_rows) {
    const int wave  = (blockIdx.x * blockDim.x + threadIdx.x) >> 5;
    const int lane  = threadIdx.x & 31;
    const int lo    = lane & 15;       // row within tile (A) / col within tile (B,C)
    const int hi    = lane >> 4;       // K-pair select
    const int strip = wave;            // 16-row strip index
    if (strip * 16 >= n_rows) return;

    const int row = strip * 16 + lo;
    const float2* __restrict__ Xrow = reinterpret_cast<const float2*>(X + (size_t)row * D);

    // Seed accumulators with the bias (constant per output column == per lane).
    const float bi0 = bias[lo +  0];
    const float bi1 = bias[lo + 16];
    const float bi2 = bias[lo + 32];
    const float bi3 = bias[lo + 48];
    v8f acc0, acc1, acc2, acc3;
#pragma unroll
    for (int i = 0; i < 8; ++i) {
        acc0[i] = bi0; acc1[i] = bi1; acc2[i] = bi2; acc3[i] = bi3;
    }

    for (int k0 = 0; k0 < D; k0 += 4) {
        // A tile 16x4: lanes 0-15 -> K=k0,k0+1 ; lanes 16-31 -> K=k0+2,k0+3
        float2 av = Xrow[(k0 >> 1) + hi];
        v2f a;
        if (RELU_IN) {
            a.x = fmaxf(av.x, 0.f);
            a.y = fmaxf(av.y, 0.f);
        } else {
            a.x = av.x;
            a.y = av.y;
        }

        const int kA = k0 + 2 * hi;
        const float* __restrict__ Wr0 = W + (size_t)kA * D;
        const float* __restrict__ Wr1 = Wr0 + D;

        v2f b0; b0.x = Wr0[lo +  0]; b0.y = Wr1[lo +  0];
        v2f b1; b1.x = Wr0[lo + 16]; b1.y = Wr1[lo + 16];
        v2f b2; b2.x = Wr0[lo + 32]; b2.y = Wr1[lo + 32];
        v2f b3; b3.x = Wr0[lo + 48]; b3.y = Wr1[lo + 48];

        acc0 = __builtin_amdgcn_wmma_f32_16x16x4_f32(false, a, false, b0, (short)0, acc0, false, false);
        acc1 = __builtin_amdgcn_wmma_f32_16x16x4_f32(false, a, false, b1, (short)0, acc1, false, false);
        acc2 = __builtin_amdgcn_wmma_f32_16x16x4_f32(false, a, false, b2, (short)0, acc2, false, false);
        acc3 = __builtin_amdgcn_wmma_f32_16x16x4_f32(false, a, false, b3, (short)0, acc3, false, false);
    }

    // C/D layout: VGPR i -> rows (i) lanes 0-15 and (i+8) lanes 16-31, col = lo
#pragma unroll
    for (int i = 0; i < 8; ++i) {
        const size_t orow = (size_t)(strip * 16 + i + 8 * hi) * D;
        H[orow + lo +  0] = acc0[i];
        H[orow + lo + 16] = acc1[i];
        H[orow + lo + 32] = acc2[i];
        H[orow + lo + 48] = acc3[i];
    }
}

// ---------------------------------------------------------------------------
// agg[dst] += h[src] + rel[edge_type]  (segment_sum via fp32 global atomics)
// One wave per edge; 32 lanes cover D=64 as 2 coalesced dwords each.
// ---------------------------------------------------------------------------
__global__ void rgcn_scatter_kernel(const float* __restrict__ h,
                                    const float* __restrict__ rel,
                                    const int* __restrict__ src,
                                    const int* __restrict__ dst,
                                    const int* __restrict__ etype,
                                    float* __restrict__ agg) {
    const int wave = (blockIdx.x * blockDim.x + threadIdx.x) >> 5;
    const int lane = threadIdx.x & 31;
    if (wave >= N_EDGES) return;

    const int s = src[wave];
    const int t = dst[wave];
    const int r = etype[wave];

    const float* __restrict__ hs = h   + (size_t)s * D;
    const float* __restrict__ rr = rel + (size_t)r * D;
    float* __restrict__       at = agg + (size_t)t * D;

    const float v0 = hs[lane]      + rr[lane];
    const float v1 = hs[lane + 32] + rr[lane + 32];
    unsafeAtomicAdd(at + lane,      v0);
    unsafeAtomicAdd(at + lane + 32, v1);
}

// ---------------------------------------------------------------------------
// scores[i] = sum_d relu(x2[s])*rel_final[r]*relu(x2[t]); one wave per triple.
// Per-lane float2 gathers -> coalesced global_load_b64.
// ---------------------------------------------------------------------------
__global__ void rgcn_score_kernel(const float* __restrict__ x2,
                                  const float* __restrict__ rel_final,
                                  const int* __restrict__ batch,
                                  float* __restrict__ out) {
    const int wave = (blockIdx.x * blockDim.x + threadIdx.x) >> 5;
    const int lane = threadIdx.x & 31;
    if (wave >= N_TRIPLES) return;

    const int s = batch[wave * 3 + 0];
    const int r = batch[wave * 3 + 1];
    const int t = batch[wave * 3 + 2];

    const float2 xs = reinterpret_cast<const float2*>(x2        + (size_t)s * D)[lane];
    const float2 xt = reinterpret_cast<const float2*>(x2        + (size_t)t * D)[lane];
    const float2 rf = reinterpret_cast<const float2*>(rel_final + (size_t)r * D)[lane];

    float acc = fmaxf(xs.x, 0.f) * rf.x * fmaxf(xt.x, 0.f)
              + fmaxf(xs.y, 0.f) * rf.y * fmaxf(xt.y, 0.f);

#pragma unroll
    for (int off = 16; off > 0; off >>= 1)
        acc += __shfl_xor(acc, off, 32);

    if (lane == 0) out[wave] = acc;
}

// ---------------------------------------------------------------------------
extern "C" void kernel_launch(void* const* d_in, const int* in_sizes, int n_in,
                              void* d_out, int out_size, void* d_ws, size_t ws_size,
                              hipStream_t stream) {
    const float* x         = (const float*)d_in[0];
    const int*   edge_idx  = (const int*)  d_in[1];   // [2, N_EDGES]
    const int*   edge_type = (const int*)  d_in[2];
    const int*   batch     = (const int*)  d_in[3];   // [B, K, 3]
    const float* W1        = (const float*)d_in[4];
    const float* b1        = (const float*)d_in[5];
    const float* rel1      = (const float*)d_in[6];
    const float* W2        = (const float*)d_in[7];
    const float* b2        = (const float*)d_in[8];
    const float* rel2      = (const float*)d_in[9];
    const float* rel_final = (const float*)d_in[10];
    float*       out       = (float*)d_out;

    const int* src = edge_idx;
    const int* dst = edge_idx + N_EDGES;

    float* h    = (float*)d_ws;                    // 100000*64 floats (25.6 MB)
    float* agg1 = h    + (size_t)N_NODES * D;      // 25.6 MB
    float* agg2 = agg1 + (size_t)N_NODES * D;      // 25.6 MB
    (void)ws_size; (void)n_in; (void)in_sizes; (void)out_size;

    // zero agg1+agg2 (contiguous): 2*N_NODES*D floats = 3.2M float4
    {
        const int n4 = (2 * N_NODES * D) / 4;
        rgcn_zero_kernel<<<(n4 + 255) / 256, 256, 0, stream>>>((float4*)agg1, n4);
    }

    const int strips      = N_NODES / 16;                       // 6250
    const int gemm_blocks = (strips * 32 + 255) / 256;          // 8 waves/block
    const int scat_blocks = ((size_t)N_EDGES * 32 + 255) / 256;
    const int scor_blocks = ((size_t)N_TRIPLES * 32 + 255) / 256;

    // Layer 1 (no input relu)
    rgcn_gemm_wmma<false><<<gemm_blocks, 256, 0, stream>>>(x, W1, b1, h, N_NODES);
    rgcn_scatter_kernel<<<scat_blocks, 256, 0, stream>>>(h, rel1, src, dst, edge_type, agg1);
    // Layer 2 (relu of agg1 fused into GEMM A-load)
    rgcn_gemm_wmma<true><<<gemm_blocks, 256, 0, stream>>>(agg1, W2, b2, h, N_NODES);
    rgcn_scatter_kernel<<<scat_blocks, 256, 0, stream>>>(h, rel2, src, dst, edge_type, agg2);
    // Scoring (relu of agg2 fused into gather)
    rgcn_score_kernel<<<scor_blocks, 256, 0, stream>>>(agg2, rel_final, batch, out);
}